// SpectralAttention_27453430956347
// MI455X (gfx1250) — compile-verified
//
#include <hip/hip_runtime.h>
#include <math.h>
#include <stdint.h>

typedef float v2f __attribute__((ext_vector_type(2)));
typedef float v8f __attribute__((ext_vector_type(8)));

// fp32 tensor-core MAC: D = A(16x4) * B(4x16) + C   (CDNA5 V_WMMA_F32_16X16X4_F32)
#define WMMA_F32(a, b, c) \
  __builtin_amdgcn_wmma_f32_16x16x4_f32(false, (a), false, (b), (short)0, (c), false, false)

// Problem constants (from reference)
constexpr int Bn = 2, Cc = 256, Hs = 48;     // W = D = 48 too
constexpr int KY = 16, KZR = 7;
constexpr int NT = 1792;                      // tokens = 16*16*7
constexpr int MALL = Bn * NT;                 // 3584 rows for projections
constexpr float PI2 = 6.283185307179586f;

// ---------------- twiddle tables (built on-device each launch) ----------------
__global__ void sa_tables(float* __restrict__ fz, float* __restrict__ ff) {
  int t = threadIdx.x;
  for (int i = t; i < 48 * 7; i += blockDim.x) {
    int d = i / 7, kz = i % 7;
    float s, c; sincosf(-PI2 * (float)(d * kz) / 48.f, &s, &c);
    fz[2 * i] = c; fz[2 * i + 1] = s;
  }
  for (int i = t; i < 48 * 16; i += blockDim.x) {
    int w = i / 16, k = i % 16;
    float s, c; sincosf(-PI2 * (float)(w * k) / 48.f, &s, &c);
    ff[2 * i] = c; ff[2 * i + 1] = s;
  }
}

// ---------------- forward partial DFT, stage 1: contract D (48 -> 7 cplx) ----------------
__global__ __launch_bounds__(256) void sa_fwd_d(const float* __restrict__ x,
                                                const float* __restrict__ fz,
                                                float* __restrict__ out) {
  __shared__ float xs[4][48];
  int ty = threadIdx.y, tx = threadIdx.x;
  long row = (long)blockIdx.x * 4 + ty;
  if (tx < 48) xs[ty][tx] = x[row * 48 + tx];
  __syncthreads();
  if (tx < 14) {
    int kz = tx >> 1, comp = tx & 1;
    float acc = 0.f;
    for (int d = 0; d < 48; ++d) acc += xs[ty][d] * fz[(d * 7 + kz) * 2 + comp];
    out[row * 14 + tx] = acc;
  }
}

// ---------------- stage 2: contract W (48 -> 16), complex in * complex twiddle ----------------
__global__ __launch_bounds__(128) void sa_fwd_w(const float* __restrict__ in,
                                                const float* __restrict__ ff,
                                                float* __restrict__ out) {
  __shared__ float s[48 * 14];
  long rowH = blockIdx.x;
  const float* ip = in + rowH * 672;
  for (int i = threadIdx.x; i < 672; i += 128) s[i] = ip[i];
  __syncthreads();
  int t = threadIdx.x;
  if (t < 112) {
    int ky = t / 7, kz = t % 7;
    float ar = 0.f, ai = 0.f;
    for (int w = 0; w < 48; ++w) {
      float xr = s[(w * 7 + kz) * 2], xi = s[(w * 7 + kz) * 2 + 1];
      float cr = ff[(w * 16 + ky) * 2], ci = ff[(w * 16 + ky) * 2 + 1];
      ar += xr * cr - xi * ci;
      ai += xr * ci + xi * cr;
    }
    out[(rowH * 112 + t) * 2]     = ar;
    out[(rowH * 112 + t) * 2 + 1] = ai;
  }
}

// ---------------- stage 3: contract H (48 -> 16) -> planar xf[b][n][c] ----------------
// LDS fill uses the CDNA5 async global->LDS path (ASYNCcnt) when compiling for gfx1250.
__global__ __launch_bounds__(256) void sa_fwd_h(const float* __restrict__ in,
                                                const float* __restrict__ ff,
                                                float* __restrict__ xre,
                                                float* __restrict__ xim) {
  __shared__ float s[48 * 112 * 2];             // 42 KB
  long bc = blockIdx.x;                          // b*256 + c
  const float* ip = in + bc * 10752;
#if defined(__gfx1250__)
  for (int i = threadIdx.x; i < 10752; i += 256) {
    unsigned lo = (unsigned)(size_t)(&s[i]);     // low 32 bits == LDS offset
    const float* g = ip + i;
    asm volatile("global_load_async_to_lds_b32 %0, %1, off" :: "v"(lo), "v"(g) : "memory");
  }
  asm volatile("s_wait_asynccnt 0x0" ::: "memory");
  __syncthreads();
#else
  for (int i = threadIdx.x; i < 10752; i += 256) s[i] = ip[i];
  __syncthreads();
#endif
  int b = (int)(bc >> 8), c = (int)(bc & 255);
  for (int o = threadIdx.x; o < 1792; o += 256) {  // o == n = kx*112 + ky*7 + kz
    int kx = o / 112, t = o % 112;
    float ar = 0.f, ai = 0.f;
    for (int h = 0; h < 48; ++h) {
      float xr = s[(h * 112 + t) * 2], xi = s[(h * 112 + t) * 2 + 1];
      float cr = ff[(h * 16 + kx) * 2], ci = ff[(h * 16 + kx) * 2 + 1];
      ar += xr * cr - xi * ci;
      ai += xr * ci + xi * cr;
    }
    xre[((long)(b * NT + o)) * 256 + c] = ar;
    xim[((long)(b * NT + o)) * 256 + c] = ai;
  }
}

// ---------------- complex-in / real-weight projection: O = A @ W^T + bias ----------------
// Register double-buffered K loop: load step k+4 while WMMAs consume step k.
__global__ __launch_bounds__(256) void sa_proj(const float* __restrict__ Ar,
                                               const float* __restrict__ Ai,
                                               const float* __restrict__ W,
                                               const float* __restrict__ bias,
                                               float* __restrict__ Or,
                                               float* __restrict__ Oi) {
  int wave = threadIdx.x >> 5, lane = threadIdx.x & 31;
  int tile = blockIdx.x * 8 + wave;             // 3584 tiles total
  int tm = tile >> 4, tn = tile & 15;
  int r = lane & 15, half = lane >> 4;
  const float* arp = Ar + (long)(tm * 16 + r) * 256 + half * 2;
  const float* aip = Ai + (long)(tm * 16 + r) * 256 + half * 2;
  const float* wp  = W  + (long)(tn * 16 + r) * 256 + half * 2;
  v8f accR = {}; v8f accI = {};
  v2f ca = {arp[0], arp[1]};
  v2f ci = {aip[0], aip[1]};
  v2f cb = {wp[0],  wp[1]};
#pragma unroll
  for (int k = 0; k < 256; k += 4) {
    int kn = (k + 4) & 255;                     // wraps on last step (dummy reload of k=0)
    v2f na = {arp[kn], arp[kn + 1]};
    v2f ni = {aip[kn], aip[kn + 1]};
    v2f nb = {wp[kn],  wp[kn + 1]};
    accR = WMMA_F32(ca, cb, accR);
    accI = WMMA_F32(ci, cb, accI);
    ca = na; ci = ni; cb = nb;
  }
  float bv = bias[tn * 16 + r];
  long base = (long)(tm * 16) * 256 + tn * 16 + r;
  for (int v = 0; v < 8; ++v) {
    int m = v + half * 8;
    Or[base + (long)m * 256] = accR[v] + bv;
    Oi[base + (long)m * 256] = accI[v];
  }
}

// ---------------- scores for one (b,h): S = (q @ k^T) / sqrt(32), complex ----------------
__global__ __launch_bounds__(256) void sa_scores(const float* __restrict__ Qr,
                                                 const float* __restrict__ Qi,
                                                 const float* __restrict__ Kr,
                                                 const float* __restrict__ Ki,
                                                 float* __restrict__ Sr,
                                                 float* __restrict__ Si) {
  int wave = threadIdx.x >> 5, lane = threadIdx.x & 31;
  int tile = blockIdx.x * 8 + wave;             // 112*112 tiles
  int tm = tile / 112, tn = tile % 112;
  int r = lane & 15, half = lane >> 4;
  const float* qrp = Qr + (long)(tm * 16 + r) * 256 + half * 2;
  const float* qip = Qi + (long)(tm * 16 + r) * 256 + half * 2;
  const float* krp = Kr + (long)(tn * 16 + r) * 256 + half * 2;
  const float* kip = Ki + (long)(tn * 16 + r) * 256 + half * 2;
  v8f rr = {}, ii = {}, ri = {}, ir = {};
  v2f car = {qrp[0], qrp[1]}, cai = {qip[0], qip[1]};
  v2f cbr = {krp[0], krp[1]}, cbi = {kip[0], kip[1]};
#pragma unroll
  for (int k = 0; k < 32; k += 4) {
    int kn = (k + 4) & 31;
    v2f nar = {qrp[kn], qrp[kn + 1]}, nai = {qip[kn], qip[kn + 1]};
    v2f nbr = {krp[kn], krp[kn + 1]}, nbi = {kip[kn], kip[kn + 1]};
    rr = WMMA_F32(car, cbr, rr);
    ii = WMMA_F32(cai, cbi, ii);
    ri = WMMA_F32(car, cbi, ri);
    ir = WMMA_F32(cai, cbr, ir);
    car = nar; cai = nai; cbr = nbr; cbi = nbi;
  }
  const float sc = 0.17677669529663688f;        // 32^-0.5
  long base = (long)(tm * 16) * NT + tn * 16 + r;
  for (int v = 0; v < 8; ++v) {
    int m = v + half * 8;
    Sr[base + (long)m * NT] = (rr[v] - ii[v]) * sc;
    Si[base + (long)m * NT] = (ri[v] + ir[v]) * sc;
  }
}

// ---------------- complex softmax (row-wise, in place) ----------------
__global__ __launch_bounds__(256) void sa_softmax(float* __restrict__ Sr,
                                                  float* __restrict__ Si) {
  __shared__ float redA[256], redB[256];
  int tid = threadIdx.x;
  float* sr = Sr + (long)blockIdx.x * NT;
  float* si = Si + (long)blockIdx.x * NT;
  float mx = -3.4e38f;
  for (int j = tid; j < NT; j += 256) mx = fmaxf(mx, sr[j]);
  redA[tid] = mx; __syncthreads();
  for (int s = 128; s > 0; s >>= 1) { if (tid < s) redA[tid] = fmaxf(redA[tid], redA[tid + s]); __syncthreads(); }
  mx = redA[0]; __syncthreads();
  float sumr = 0.f, sumi = 0.f;
  for (int j = tid; j < NT; j += 256) {
    float er = __expf(sr[j] - mx);
    float sn, cs; __sincosf(si[j], &sn, &cs);
    float xr = er * cs, xi = er * sn;
    sr[j] = xr; si[j] = xi;
    sumr += xr; sumi += xi;
  }
  redA[tid] = sumr; redB[tid] = sumi; __syncthreads();
  for (int s = 128; s > 0; s >>= 1) { if (tid < s) { redA[tid] += redA[tid + s]; redB[tid] += redB[tid + s]; } __syncthreads(); }
  sumr = redA[0]; sumi = redB[0];
  float inv = 1.f / (sumr * sumr + sumi * sumi);   // e / S = e*conj(S)/|S|^2
  for (int j = tid; j < NT; j += 256) {
    float xr = sr[j], xi = si[j];
    sr[j] = (xr * sumr + xi * sumi) * inv;
    si[j] = (xi * sumr - xr * sumi) * inv;
  }
}

// ---------------- transpose V planes: [b][n][256] -> [b][256][n] (for A·V B-operand) ----------------
__global__ __launch_bounds__(256) void sa_transpose_v(const float* __restrict__ Vr,
                                                      const float* __restrict__ Vi,
                                                      float* __restrict__ Tr,
                                                      float* __restrict__ Ti) {
  __shared__ float tr[32][33], ti[32][33];
  int b = blockIdx.z;
  int n0 = blockIdx.x * 32, c0 = blockIdx.y * 32;
  int tx = threadIdx.x & 31, ty = threadIdx.x >> 5;   // 32 x 8
  for (int i = 0; i < 32; i += 8) {
    long src = ((long)(b * NT + n0 + ty + i)) * 256 + c0 + tx;
    tr[ty + i][tx] = Vr[src];
    ti[ty + i][tx] = Vi[src];
  }
  __syncthreads();
  for (int i = 0; i < 32; i += 8) {
    long dst = ((long)(b * 256 + c0 + ty + i)) * NT + n0 + tx;
    Tr[dst] = tr[tx][ty + i];
    Ti[dst] = ti[tx][ty + i];
  }
}

// ---------------- O = attn @ v for one (b,h): M=1792, N=32, K=1792, complex ----------------
// V pre-transposed => B-operand is row-major N x K (contiguous b64 fragment loads).
__global__ __launch_bounds__(256) void sa_av(const float* __restrict__ Ar,
                                             const float* __restrict__ Ai,
                                             const float* __restrict__ Vtr,
                                             const float* __restrict__ Vti,
                                             float* __restrict__ Or,
                                             float* __restrict__ Oi) {
  int wave = threadIdx.x >> 5, lane = threadIdx.x & 31;
  int tile = blockIdx.x * 8 + wave;             // 224 tiles: 112 x 2
  int tm = tile >> 1, tn = tile & 1;
  int r = lane & 15, half = lane >> 4;
  const float* arp = Ar  + (long)(tm * 16 + r) * NT + half * 2;
  const float* aip = Ai  + (long)(tm * 16 + r) * NT + half * 2;
  const float* vrp = Vtr + (long)(tn * 16 + r) * NT + half * 2;
  const float* vip = Vti + (long)(tn * 16 + r) * NT + half * 2;
  v8f rr = {}, ii = {}, ri = {}, ir = {};
  v2f car = {arp[0], arp[1]}, cai = {aip[0], aip[1]};
  v2f cbr = {vrp[0], vrp[1]}, cbi = {vip[0], vip[1]};
#pragma unroll 8
  for (int k = 0; k < NT; k += 4) {
    int kn = (k + 4 < NT) ? (k + 4) : 0;        // uniform select, EXEC stays full
    __builtin_prefetch(arp + k + 256, 0, 1);    // global_prefetch_b8 down the K-stream
    v2f nar = {arp[kn], arp[kn + 1]}, nai = {aip[kn], aip[kn + 1]};
    v2f nbr = {vrp[kn], vrp[kn + 1]}, nbi = {vip[kn], vip[kn + 1]};
    rr = WMMA_F32(car, cbr, rr);
    ii = WMMA_F32(cai, cbi, ii);
    ri = WMMA_F32(car, cbi, ri);
    ir = WMMA_F32(cai, cbr, ir);
    car = nar; cai = nai; cbr = nbr; cbi = nbi;
  }
  long base = (long)(tm * 16) * 256 + tn * 16 + r;
  for (int v = 0; v < 8; ++v) {
    int m = v + half * 8;
    Or[base + (long)m * 256] = rr[v] - ii[v];
    Oi[base + (long)m * 256] = ri[v] + ir[v];
  }
}

// ---------------- inverse: ifft along H (16 -> 48), per (b,c) ----------------
__global__ __launch_bounds__(256) void sa_inv_h(const float* __restrict__ Pr,
                                                const float* __restrict__ Pi,
                                                const float* __restrict__ ff,
                                                float* __restrict__ out) {
  __shared__ float sr[1792], sii[1792];
  long bc = blockIdx.x;
  int b = (int)(bc >> 8), c = (int)(bc & 255);
  for (int nn = threadIdx.x; nn < 1792; nn += 256) {
    sr[nn]  = Pr[((long)(b * NT + nn)) * 256 + c];
    sii[nn] = Pi[((long)(b * NT + nn)) * 256 + c];
  }
  __syncthreads();
  for (int o = threadIdx.x; o < 48 * 112; o += 256) {
    int h = o / 112, t = o % 112;
    float ar = 0.f, ai = 0.f;
    for (int kx = 0; kx < 16; ++kx) {
      float xr = sr[kx * 112 + t], xi = sii[kx * 112 + t];
      float cr = ff[(h * 16 + kx) * 2], ci = -ff[(h * 16 + kx) * 2 + 1];  // conj -> e^{+i}
      ar += xr * cr - xi * ci;
      ai += xr * ci + xi * cr;
    }
    out[(bc * 5376 + o) * 2]     = ar;
    out[(bc * 5376 + o) * 2 + 1] = ai;
  }
}

// ---------------- inverse: ifft along W (16 -> 48), per (b,c,h) ----------------
__global__ __launch_bounds__(128) void sa_inv_w(const float* __restrict__ in,
                                                const float* __restrict__ ff,
                                                float* __restrict__ out) {
  __shared__ float s[224];
  long rowH = blockIdx.x;
  const float* ip = in + rowH * 224;
  for (int i = threadIdx.x; i < 224; i += 128) s[i] = ip[i];
  __syncthreads();
  for (int o = threadIdx.x; o < 336; o += 128) {
    int w = o / 7, kz = o % 7;
    float ar = 0.f, ai = 0.f;
    for (int ky = 0; ky < 16; ++ky) {
      float xr = s[(ky * 7 + kz) * 2], xi = s[(ky * 7 + kz) * 2 + 1];
      float cr = ff[(w * 16 + ky) * 2], ci = -ff[(w * 16 + ky) * 2 + 1];
      ar += xr * cr - xi * ci;
      ai += xr * ci + xi * cr;
    }
    out[(rowH * 336 + o) * 2]     = ar;
    out[(rowH * 336 + o) * 2 + 1] = ai;
  }
}

// ---------------- inverse: irfft along D (7 half-spectrum -> 48 real) + 1/48^3 ----------------
__global__ __launch_bounds__(256) void sa_inv_d(const float* __restrict__ in,
                                                const float* __restrict__ fz,
                                                float* __restrict__ y) {
  __shared__ float s[4][14];
  int ty = threadIdx.y, d = threadIdx.x;
  long row = (long)blockIdx.x * 4 + ty;
  if (d < 14) s[ty][d] = in[row * 14 + d];
  __syncthreads();
  if (d < 48) {
    float acc = s[ty][0];                        // kz=0: Re(X0)
    for (int kz = 1; kz < 7; ++kz) {             // 2*Re(X e^{+i th}) via conj of forward table
      float xr = s[ty][2 * kz], xi = s[ty][2 * kz + 1];
      acc += 2.f * (xr * fz[(d * 7 + kz) * 2] + xi * fz[(d * 7 + kz) * 2 + 1]);
    }
    y[row * 48 + d] = acc * (1.f / 110592.f);    // 1/48^3
  }
}

// ======================= host launcher =======================
extern "C" void kernel_launch(void* const* d_in, const int* in_sizes, int n_in,
                              void* d_out, int out_size, void* d_ws, size_t ws_size,
                              hipStream_t stream) {
  (void)in_sizes; (void)n_in; (void)out_size; (void)ws_size;
  const float* x  = (const float*)d_in[0];
  const float* wq = (const float*)d_in[1];
  const float* bq = (const float*)d_in[2];
  const float* wk = (const float*)d_in[3];
  const float* bk = (const float*)d_in[4];
  const float* wv = (const float*)d_in[5];
  const float* bv = (const float*)d_in[6];
  const float* wo = (const float*)d_in[7];
  const float* bo = (const float*)d_in[8];
  float* out = (float*)d_out;
  float* ws  = (float*)d_ws;

  size_t off = 0;
  auto alloc = [&](size_t n) { size_t r = off; off += (n + 255) & ~(size_t)255; return r; };
  float* fz   = ws + alloc(48 * 7 * 2);
  float* ff   = ws + alloc(48 * 16 * 2);
  float* buf1 = ws + alloc((size_t)Bn * Cc * Hs * 48 * KZR * 2);   // 66 MB; reused as t2
  float* buf2 = ws + alloc((size_t)Bn * Cc * Hs * KY * KZR * 2);   // 22 MB; reused as t1
  float* xre  = ws + alloc((size_t)MALL * 256);
  float* xim  = ws + alloc((size_t)MALL * 256);
  float* qre  = ws + alloc((size_t)MALL * 256);
  float* qim  = ws + alloc((size_t)MALL * 256);
  float* kre  = ws + alloc((size_t)MALL * 256);
  float* kim  = ws + alloc((size_t)MALL * 256);
  float* vre  = ws + alloc((size_t)MALL * 256);
  float* vim  = ws + alloc((size_t)MALL * 256);
  float* vtr  = ws + alloc((size_t)MALL * 256);                     // V transposed (per b: 256 x NT)
  float* vti  = ws + alloc((size_t)MALL * 256);
  float* asr  = ws + alloc((size_t)NT * NT);                        // one (b,h) slice, reused 16x
  float* asi  = ws + alloc((size_t)NT * NT);
  float* ore  = ws + alloc((size_t)MALL * 256);
  float* oim  = ws + alloc((size_t)MALL * 256);
  float* pre  = ws + alloc((size_t)MALL * 256);
  float* pim  = ws + alloc((size_t)MALL * 256);

  const int ROWS = Bn * Cc * Hs * 48;           // 1,179,648 (b,c,h,w)
  dim3 blk64x4(64, 4);

  sa_tables<<<1, 256, 0, stream>>>(fz, ff);
  sa_fwd_d<<<ROWS / 4, blk64x4, 0, stream>>>(x, fz, buf1);
  sa_fwd_w<<<Bn * Cc * Hs, 128, 0, stream>>>(buf1, ff, buf2);
  sa_fwd_h<<<Bn * Cc, 256, 0, stream>>>(buf2, ff, xre, xim);

  sa_proj<<<448, 256, 0, stream>>>(xre, xim, wq, bq, qre, qim);
  sa_proj<<<448, 256, 0, stream>>>(xre, xim, wk, bk, kre, kim);
  sa_proj<<<448, 256, 0, stream>>>(xre, xim, wv, bv, vre, vim);

  dim3 tgrid(NT / 32, 256 / 32, Bn);
  sa_transpose_v<<<tgrid, 256, 0, stream>>>(vre, vim, vtr, vti);

  for (int b = 0; b < Bn; ++b) {
    for (int h = 0; h < 8; ++h) {
      long qo = (long)b * NT * 256 + (long)h * 32;          // q/k row-major sub-plane
      long vo = ((long)b * 256 + (long)h * 32) * NT;        // transposed V sub-plane
      sa_scores<<<(112 * 112) / 8, 256, 0, stream>>>(qre + qo, qim + qo, kre + qo, kim + qo, asr, asi);
      sa_softmax<<<NT, 256, 0, stream>>>(asr, asi);
      sa_av<<<224 / 8, 256, 0, stream>>>(asr, asi, vtr + vo, vti + vo, ore + qo, oim + qo);
    }
  }

  sa_proj<<<448, 256, 0, stream>>>(ore, oim, wo, bo, pre, pim);

  sa_inv_h<<<Bn * Cc, 256, 0, stream>>>(pre, pim, ff, buf2);
  sa_inv_w<<<Bn * Cc * Hs, 128, 0, stream>>>(buf2, ff, buf1);
  sa_inv_d<<<ROWS / 4, blk64x4, 0, stream>>>(buf1, fz, out);
}